// LatentGraph_43052752175368
// MI455X (gfx1250) — compile-verified
//
#include <hip/hip_runtime.h>

// ---------------------------------------------------------------------------
// LatentGraph forward for MI455X (gfx1250, wave32).
// fp32 end-to-end; dense GEMMs use V_WMMA_F32_16X16X4_F32 with the B matrix
// staged into LDS by the Tensor Data Mover (tensor_load_to_lds + TENSORcnt);
// irregular scatter uses hardware global_atomic_add_f32 (L2-resident).
// ---------------------------------------------------------------------------

namespace lg {
constexpr int N  = 50000;   // nodes
constexpr int E  = 800000;  // edges
constexpr int DN = 128;     // node in-features
constexpr int DE = 32;      // edge in-features
constexpr int LE = 16;      // latent edge channels
constexpr int LN = 64;      // latent node features
constexpr int DO = 128;     // out features
}

typedef __attribute__((ext_vector_type(2))) float v2f;
typedef __attribute__((ext_vector_type(8))) float v8f;
typedef __attribute__((ext_vector_type(4))) unsigned int v4u;
typedef __attribute__((ext_vector_type(4))) int v4i;
typedef __attribute__((ext_vector_type(8))) int v8i;

#if defined(__has_builtin)
#if __has_builtin(__builtin_amdgcn_tensor_load_to_lds) && \
    __has_builtin(__builtin_amdgcn_s_wait_tensorcnt)
#define LG_USE_TDM 1
#endif
#endif
#ifndef LG_USE_TDM
#define LG_USE_TDM 0
#endif

// Non-returning hardware float atomic (global_atomic_add_f32, STOREcnt).
__device__ __forceinline__ void atomic_add_f32(float* p, float v) {
  __hip_atomic_fetch_add(p, v, __ATOMIC_RELAXED, __HIP_MEMORY_SCOPE_AGENT);
}

#if LG_USE_TDM
// TDM: DMA a 2-D row-major f32 tile (dim1 rows x dim0 elems, row stride
// stride0 elems) from global into LDS at byte offset lds_off.
// D# layout per CDNA5 ISA (groups 2/3 zero => 2-D).
// 6-arg builtin form (clang-23 / amdgpu-toolchain):
//   (v4u g0, v8i g1, v4i g2, v4i g3, v8i g4, i32 cpol)
__device__ __forceinline__ void tdm_load_2d_to_lds(const float* gsrc,
                                                   unsigned lds_off,
                                                   int dim0, int dim1,
                                                   int stride0) {
  const unsigned long long ga = (unsigned long long)(size_t)gsrc;
  v4u g0;
  g0.x = 1u;                                        // count=1, user mode
  g0.y = lds_off;                                   // lds_addr (bytes)
  g0.z = (unsigned)(ga & 0xffffffffu);              // global_addr[31:0]
  g0.w = (unsigned)((ga >> 32) & 0x1ffffffu)        // global_addr[56:32]
         | 0x80000000u;                             // type=2 ("image")
  v8i g1;
  g1[0] = (2 << 16);                                // wg_mask=0, data_size=4B
  g1[1] = (dim0 & 0xffff) << 16;                    // tensor_dim0[15:0]
  g1[2] = ((dim0 >> 16) & 0xffff) |                 // tensor_dim0[31:16]
          ((dim1 & 0xffff) << 16);                  // tensor_dim1[15:0]
  g1[3] = ((dim1 >> 16) & 0xffff) |                 // tensor_dim1[31:16]
          ((dim0 & 0xffff) << 16);                  // tile_dim0 = dim0
  g1[4] = dim1 & 0xffff;                            // tile_dim1 = dim1, tile_dim2=0
  g1[5] = stride0;                                  // tensor_dim0_stride[31:0]
  g1[6] = 0;                                        // stride hi, dim1_stride lo
  g1[7] = 0;
  const v4i gz4 = {0, 0, 0, 0};
  const v8i gz8 = {0, 0, 0, 0, 0, 0, 0, 0};
  __builtin_amdgcn_tensor_load_to_lds(g0, g1, gz4, gz4, gz8, 0);
}
#endif

// Stage B[dim1 x dim0] (row-major, stride0 == dim0) into LDS, whole block.
__device__ __forceinline__ void stage_B_to_lds(const float* __restrict__ B,
                                               float* sB, int elems,
                                               int dim0, int dim1) {
#if LG_USE_TDM
  if (threadIdx.x < 32) {                  // one wave issues; TDM ignores EXEC
    tdm_load_2d_to_lds(B, (unsigned)(size_t)sB, dim0, dim1, dim0);
    __builtin_amdgcn_s_wait_tensorcnt(0);  // s_wait_tensorcnt 0
  }
  __syncthreads();
#else
  for (int i = threadIdx.x; i < elems; i += blockDim.x) sB[i] = B[i];
  __syncthreads();
#endif
}

// One wave computes C[16x16] += A[16xK] * B[Kx16] with V_WMMA_F32_16X16X4_F32.
// ISA fragment layout (CDNA5 ISA):
//   A 16x4 : lanes 0-15 -> M=lane, VGPR r -> K = r + 2*(lane>=16)
//   B 4x16 : lane%16 -> N, VGPR r -> K = r + 2*(lane>=16)
//   C 16x16: lane%16 -> N, VGPR r -> M = r + 8*(lane>=16)
template <int K>
__device__ __forceinline__ v8f wmma_f32_tile(const float* __restrict__ A, int lda,
                                             const float* B, int ldb, v8f c) {
  const int lane = threadIdx.x & 31;
  const int hi   = lane >> 4;   // 0 or 1
  const int lm   = lane & 15;
#pragma unroll
  for (int k0 = 0; k0 < K; k0 += 4) {
    const int ka = k0 + 2 * hi;
    v2f a, b;
    a.x = A[lm * lda + ka];
    a.y = A[lm * lda + ka + 1];
    b.x = B[ka * ldb + lm];
    b.y = B[(ka + 1) * ldb + lm];
    c = __builtin_amdgcn_wmma_f32_16x16x4_f32(false, a, false, b, (short)0, c,
                                              false, false);
  }
  return c;
}

// --- init: zero output accumulator, degree = 1.0 (self loop weight) ---------
__global__ void k_init(float* __restrict__ Y, float* __restrict__ deg) {
  const int t = blockIdx.x * blockDim.x + threadIdx.x;
  if (t < lg::N * lg::DO) Y[t] = 0.0f;
  if (t < lg::N * lg::LE) deg[t] = 1.0f;
}

// --- narrow int64 edge indices to int32 once --------------------------------
__global__ void k_idx(const long long* __restrict__ ei, int* __restrict__ row32,
                      int* __restrict__ col32) {
  const int e = blockIdx.x * blockDim.x + threadIdx.x;
  if (e < lg::E) {
    row32[e] = (int)ei[e];            // source
    col32[e] = (int)ei[lg::E + e];    // target (aggregation)
  }
}

// --- edge MLP: w = relu(edge_attr @ W_edge + b_edge); deg[col] += w ---------
__global__ void k_edge(const float* __restrict__ ea, const float* __restrict__ We,
                       const float* __restrict__ be, const int* __restrict__ col32,
                       float* __restrict__ w, float* __restrict__ deg) {
  __shared__ float sW[lg::DE * lg::LE];
  __shared__ float sb[lg::LE];
  for (int i = threadIdx.x; i < lg::DE * lg::LE; i += blockDim.x) sW[i] = We[i];
  if (threadIdx.x < lg::LE) sb[threadIdx.x] = be[threadIdx.x];
  __syncthreads();
  const int e = blockIdx.x * blockDim.x + threadIdx.x;
  if (e >= lg::E) return;
  float acc[lg::LE];
#pragma unroll
  for (int l = 0; l < lg::LE; ++l) acc[l] = sb[l];
  const float* a = ea + (size_t)e * lg::DE;
#pragma unroll
  for (int j = 0; j < lg::DE; ++j) {
    const float aj = a[j];
#pragma unroll
    for (int l = 0; l < lg::LE; ++l) acc[l] += aj * sW[j * lg::LE + l];
  }
  const int c = col32[e];
#pragma unroll
  for (int l = 0; l < lg::LE; ++l) {
    const float v = acc[l] > 0.0f ? acc[l] : 0.0f;
    w[(size_t)e * lg::LE + l] = v;
    atomic_add_f32(&deg[c * lg::LE + l], v);
  }
}

// --- xW = x @ W_gcn  ([N,128]x[128,64]) via WMMA; 4 waves = 4 col tiles -----
__global__ void k_xw(const float* __restrict__ x, const float* __restrict__ Wg,
                     float* __restrict__ xW) {
  __shared__ float sB[lg::DN * lg::LN];            // 32 KB
  stage_B_to_lds(Wg, sB, lg::DN * lg::LN, lg::LN, lg::DN);
  const int wave = threadIdx.x >> 5;               // 0..3
  const int row0 = blockIdx.x * 16;
  const int col0 = wave * 16;
  v8f c = {};
  c = wmma_f32_tile<lg::DN>(x + (size_t)row0 * lg::DN, lg::DN, sB + col0, lg::LN, c);
  const int lane = threadIdx.x & 31;
  const int hi = lane >> 4, lm = lane & 15;
#pragma unroll
  for (int r = 0; r < 8; ++r)
    xW[(size_t)(row0 + r + 8 * hi) * lg::LN + col0 + lm] = c[r];
}

// --- deg -> dinv = rsqrt(deg) (in place) ------------------------------------
__global__ void k_dinv(float* __restrict__ deg) {
  const int t = blockIdx.x * blockDim.x + threadIdx.x;
  if (t < lg::N * lg::LE) {
    const float v = deg[t];
    deg[t] = v > 0.0f ? rsqrtf(v) : 0.0f;
  }
}

// --- out_l init: self-loop term + b_gcn -------------------------------------
__global__ void k_init_out(const float* __restrict__ xW, const float* __restrict__ dinv,
                           const float* __restrict__ bg, int l, float* __restrict__ out) {
  const int t = blockIdx.x * blockDim.x + threadIdx.x;
  if (t >= lg::N * lg::LN) return;
  const int n = t >> 6, f = t & 63;
  const float d = dinv[n * lg::LE + l];
  out[t] = bg[f] + d * d * xW[t];
}

// --- per-edge normalization coefficient for channel l -----------------------
__global__ void k_coef(const float* __restrict__ w, const float* __restrict__ dinv,
                       const int* __restrict__ row32, const int* __restrict__ col32,
                       int l, float* __restrict__ coef) {
  const int e = blockIdx.x * blockDim.x + threadIdx.x;
  if (e < lg::E)
    coef[e] = dinv[row32[e] * lg::LE + l] * w[(size_t)e * lg::LE + l] *
              dinv[col32[e] * lg::LE + l];
}

// --- scatter: out[col] += coef * xW[row]; 64 threads per edge ---------------
__global__ void k_scatter(const float* __restrict__ xW, const float* __restrict__ coef,
                          const int* __restrict__ row32, const int* __restrict__ col32,
                          float* __restrict__ out) {
  const int t = blockIdx.x * blockDim.x + threadIdx.x;   // exactly E*64 threads
  const int e = t >> 6, f = t & 63;
  const float c = coef[e];
  if (c != 0.0f)
    atomic_add_f32(&out[(size_t)col32[e] * lg::LN + f],
                   c * xW[(size_t)row32[e] * lg::LN + f]);
}

// --- Y += out_l @ W_node[l*64:(l+1)*64, :] via WMMA; 8 waves = 8 col tiles --
__global__ void k_gemm_acc(const float* __restrict__ A,   // [N,64]
                           const float* __restrict__ B,   // [64,128] slice of W_node
                           float* __restrict__ Y) {       // [N,128]
  __shared__ float sB[lg::LN * lg::DO];            // 32 KB
  stage_B_to_lds(B, sB, lg::LN * lg::DO, lg::DO, lg::LN);
  const int wave = threadIdx.x >> 5;               // 0..7
  const int row0 = blockIdx.x * 16;
  const int col0 = wave * 16;
  const int lane = threadIdx.x & 31;
  const int hi = lane >> 4, lm = lane & 15;
  v8f c;
#pragma unroll
  for (int r = 0; r < 8; ++r)
    c[r] = Y[(size_t)(row0 + r + 8 * hi) * lg::DO + col0 + lm];
  c = wmma_f32_tile<lg::LN>(A + (size_t)row0 * lg::LN, lg::LN, sB + col0, lg::DO, c);
#pragma unroll
  for (int r = 0; r < 8; ++r)
    Y[(size_t)(row0 + r + 8 * hi) * lg::DO + col0 + lm] = c[r];
}

// --- finalize: Y = relu(Y + b_node) -----------------------------------------
__global__ void k_final(float* __restrict__ Y, const float* __restrict__ bn) {
  const int t = blockIdx.x * blockDim.x + threadIdx.x;
  if (t >= lg::N * lg::DO) return;
  const float v = Y[t] + bn[t & (lg::DO - 1)];
  Y[t] = v > 0.0f ? v : 0.0f;
}

extern "C" void kernel_launch(void* const* d_in, const int* in_sizes, int n_in,
                              void* d_out, int out_size, void* d_ws, size_t ws_size,
                              hipStream_t stream) {
  (void)in_sizes; (void)n_in; (void)out_size; (void)ws_size;
  const float*     x  = (const float*)d_in[0];
  const long long* ei = (const long long*)d_in[1];   // int64 per reference
  const float*     ea = (const float*)d_in[2];
  const float*     Wg = (const float*)d_in[3];
  const float*     bg = (const float*)d_in[4];
  const float*     We = (const float*)d_in[5];
  const float*     be = (const float*)d_in[6];
  const float*     Wn = (const float*)d_in[7];
  const float*     bn = (const float*)d_in[8];
  float* Y = (float*)d_out;

  // Workspace layout (~90 MB total)
  char* p = (char*)d_ws;
  float* xW    = (float*)p;  p += (size_t)lg::N * lg::LN * sizeof(float);  // 12.8 MB
  float* w     = (float*)p;  p += (size_t)lg::E * lg::LE * sizeof(float);  // 51.2 MB
  float* dinv  = (float*)p;  p += (size_t)lg::N * lg::LE * sizeof(float);  //  3.2 MB (deg, then dinv)
  float* outl  = (float*)p;  p += (size_t)lg::N * lg::LN * sizeof(float);  // 12.8 MB
  float* coef  = (float*)p;  p += (size_t)lg::E * sizeof(float);           //  3.2 MB
  int*   row32 = (int*)p;    p += (size_t)lg::E * sizeof(int);             //  3.2 MB
  int*   col32 = (int*)p;                                                  //  3.2 MB

  const dim3 b256(256);
  k_init<<<(lg::N * lg::DO + 255) / 256, b256, 0, stream>>>(Y, dinv);
  k_idx <<<(lg::E + 255) / 256, b256, 0, stream>>>(ei, row32, col32);
  k_edge<<<(lg::E + 255) / 256, b256, 0, stream>>>(ea, We, be, col32, w, dinv);
  k_xw  <<<lg::N / 16, dim3(128), 0, stream>>>(x, Wg, xW);
  k_dinv<<<(lg::N * lg::LE + 255) / 256, b256, 0, stream>>>(dinv);

  for (int l = 0; l < lg::LE; ++l) {
    k_init_out<<<(lg::N * lg::LN + 255) / 256, b256, 0, stream>>>(xW, dinv, bg, l, outl);
    k_coef    <<<(lg::E + 255) / 256, b256, 0, stream>>>(w, dinv, row32, col32, l, coef);
    k_scatter <<<(lg::E * lg::LN) / 256, b256, 0, stream>>>(xW, coef, row32, col32, outl);
    k_gemm_acc<<<lg::N / 16, b256, 0, stream>>>(outl, Wn + (size_t)l * lg::LN * lg::DO, Y);
  }
  k_final<<<(lg::N * lg::DO + 255) / 256, b256, 0, stream>>>(Y, bn);
}